// UniGCNConv2_50749333569737
// MI455X (gfx1250) — compile-verified
//
#include <hip/hip_runtime.h>

#define D 64        // input feature dim
#define DO 128      // output dim (8 heads * 16)

typedef __attribute__((ext_vector_type(2))) float v2f;
typedef __attribute__((ext_vector_type(8))) float v8f;

// ---------------- zero workspace ----------------
__global__ void zero_ws_kernel(float* __restrict__ p, long n) {
    long i = (long)blockIdx.x * blockDim.x + threadIdx.x;
    if (i < n) p[i] = 0.0f;
}

// ---------------- pass 1: vertex -> hyperedge scatter (sum + counts) ----------------
// thread t handles incidence pair e = t/64, feature d = t%64.
// Consecutive lanes cover one 256B feature row -> coalesced gather and coalesced atomics.
__global__ void scatter_ve_kernel(const float* __restrict__ X,
                                  const int* __restrict__ vertex,
                                  const int* __restrict__ edges,
                                  float* __restrict__ sums,
                                  float* __restrict__ counts,
                                  int E) {
    long t = (long)blockIdx.x * blockDim.x + threadIdx.x;
    if (t >= (long)E * D) return;
    int e = (int)(t >> 6);
    int d = (int)(t & 63);
    int v = vertex[e];
    int h = edges[e];
    atomicAdd(&sums[(long)h * D + d], X[(long)v * D + d]);
    if (d == 0) atomicAdd(&counts[h], 1.0f);
}

// ---------------- finalize hyperedge features: mean * dege (in place) ----------------
__global__ void finalize_edges_kernel(float* __restrict__ sums,
                                      const float* __restrict__ counts,
                                      const float* __restrict__ dege,
                                      int M) {
    long t = (long)blockIdx.x * blockDim.x + threadIdx.x;
    if (t >= (long)M * D) return;
    int h = (int)(t >> 6);
    float c = counts[h];
    float denom = c > 1.0f ? c : 1.0f;
    sums[t] = (sums[t] / denom) * dege[h];
}

// ---------------- pass 2: hyperedge -> vertex scatter (sum) ----------------
__global__ void scatter_ev_kernel(const float* __restrict__ Xe,
                                  const int* __restrict__ vertex,
                                  const int* __restrict__ edges,
                                  float* __restrict__ Xv,
                                  int E) {
    long t = (long)blockIdx.x * blockDim.x + threadIdx.x;
    if (t >= (long)E * D) return;
    int e = (int)(t >> 6);
    int d = (int)(t & 63);
    int v = vertex[e];
    int h = edges[e];
    atomicAdd(&Xv[(long)v * D + d], Xe[(long)h * D + d]);
}

// ---------------- degv scale + row L2 normalize (one wave32 per row, in place) ----------
__global__ void normalize_rows_kernel(float* __restrict__ Xv,
                                      const float* __restrict__ degv,
                                      int N) {
    int lane = threadIdx.x & 31;
    int row  = blockIdx.x * (blockDim.x >> 5) + (threadIdx.x >> 5);
    if (row >= N) return;
    float dv = degv[row];
    float a = Xv[(long)row * D + lane]      * dv;
    float b = Xv[(long)row * D + lane + 32] * dv;
    float ss = a * a + b * b;
    // wave32 butterfly reduction
    #pragma unroll
    for (int off = 16; off > 0; off >>= 1)
        ss += __shfl_xor(ss, off, 32);
    float scale = ss > 0.0f ? (1.0f / sqrtf(ss)) : 0.0f;
    Xv[(long)row * D + lane]      = a * scale;
    Xv[(long)row * D + lane + 32] = b * scale;
}

// ---------------- dense tail GEMM: out[16 rows, 128] = Xn @ W + b via fp32 WMMA -------
// Block = 256 threads = 8 waves. Block owns 16 rows; wave w owns column tile w (16 cols).
// Per-wave: D accum (16x16 f32, 8 VGPRs), K=64 consumed as 16 chunks of K=4 with
// v_wmma_f32_16x16x4_f32. No divergence inside the WMMA loop (EXEC all-ones).
__global__ void gemm_wmma_kernel(const float* __restrict__ Xn,
                                 const float* __restrict__ W,
                                 const float* __restrict__ bias,
                                 float* __restrict__ out,
                                 int N) {
    __shared__ float lA[16 * D];   // 4 KB A tile
    const int tid  = threadIdx.x;
    const int base = blockIdx.x * 16;

    // cooperative load of the 16x64 fp32 A tile (256 threads * 1 float4 = 1024 floats)
    {
        long idx = (long)base * D + (long)tid * 4;   // element index
        float4 v = make_float4(0.f, 0.f, 0.f, 0.f);
        if (idx + 3 < (long)N * D) v = *(const float4*)(Xn + idx);
        *(float4*)(lA + tid * 4) = v;
    }
    __syncthreads();

    const int wv   = tid >> 5;        // wave id = column tile 0..7
    const int lane = tid & 31;
    const int row  = lane & 15;       // A-matrix M for this lane
    const int kh   = lane >> 4;       // 0: K={0,1}, 1: K={2,3} within a K=4 chunk
    const int col  = wv * 16 + (lane & 15);

    v8f c = {};                       // 16x16 f32 accumulator (8 VGPRs)
    #pragma unroll
    for (int kc = 0; kc < 16; ++kc) {
        const int k0 = kc * 4 + kh * 2;
        v2f a, b;
        a.x = lA[row * D + k0];
        a.y = lA[row * D + k0 + 1];
        b.x = W[k0 * DO + col];       // W is row-major [64][128]
        b.y = W[(k0 + 1) * DO + col];
        // (neg_a, A, neg_b, B, c_mod, C, reuse_a, reuse_b)
        c = __builtin_amdgcn_wmma_f32_16x16x4_f32(false, a, false, b,
                                                  (short)0, c, false, false);
    }

    const float bb = bias[col];
    #pragma unroll
    for (int r = 0; r < 8; ++r) {
        const int m = r + kh * 8;     // D-matrix row for VGPR r in this lane half
        if (base + m < N)
            out[(long)(base + m) * DO + col] = c[r] + bb;
    }
}

// ---------------- launch ----------------
extern "C" void kernel_launch(void* const* d_in, const int* in_sizes, int n_in,
                              void* d_out, int out_size, void* d_ws, size_t ws_size,
                              hipStream_t stream) {
    const float* X      = (const float*)d_in[0];
    const int*   vertex = (const int*)  d_in[1];
    const int*   edges  = (const int*)  d_in[2];
    const float* dege   = (const float*)d_in[3];
    const float* degv   = (const float*)d_in[4];
    const float* W      = (const float*)d_in[5];
    const float* bias   = (const float*)d_in[6];
    float*       out    = (float*)d_out;

    const int N = in_sizes[0] / D;   // nodes
    const int E = in_sizes[1];       // incidence pairs
    const int M = in_sizes[3];       // hyperedges

    // workspace layout (contiguous): sums[M*64] | counts[M] | Xv[N*64]
    float* sums   = (float*)d_ws;
    float* counts = sums + (size_t)M * D;
    float* Xv     = counts + (size_t)M;

    const long zn = (long)M * D + (long)M + (long)N * D;
    zero_ws_kernel<<<(int)((zn + 255) / 256), 256, 0, stream>>>(sums, zn);

    const long tE = (long)E * D;
    scatter_ve_kernel<<<(int)((tE + 255) / 256), 256, 0, stream>>>(
        X, vertex, edges, sums, counts, E);

    const long tM = (long)M * D;
    finalize_edges_kernel<<<(int)((tM + 255) / 256), 256, 0, stream>>>(
        sums, counts, dege, M);

    scatter_ev_kernel<<<(int)((tE + 255) / 256), 256, 0, stream>>>(
        sums, vertex, edges, Xv, E);

    // one wave32 per row, 4 rows per 128-thread block
    normalize_rows_kernel<<<(N + 3) / 4, 128, 0, stream>>>(Xv, degv, N);

    // one 256-thread block per 16-row tile
    gemm_wmma_kernel<<<(N + 15) / 16, 256, 0, stream>>>(Xv, W, bias, out, N);
}